// RNN_22539988370034
// MI455X (gfx1250) — compile-verified
//
#include <hip/hip_runtime.h>
#include <math.h>

// ---------------------------------------------------------------------------
// Greedy LSTM decoder (image-caption style) on MI455X / gfx1250.
//   B=64, EMB=512, HID=512, VOCAB=32000, 50 steps, wave32, WMMA f32 16x16x4.
// Per step: gather -> gates GEMM (WMMA) -> cell -> logits GEMM (WMMA) ->
//           argmax -> done-flag update.  ~300 small kernels on one stream.
// ---------------------------------------------------------------------------

#define B      64
#define EMB    512
#define HID    512
#define VOCAB  32000
#define STEPS  50
#define G4H    2048          // 4*HID
#define KCAT   1024          // EMB + HID
#define OUTROW ((size_t)STEPS * VOCAB)

typedef __attribute__((ext_vector_type(2))) float v2f;
typedef __attribute__((ext_vector_type(8))) float v8f;

// ---------------------------------------------------------------------------
// WMMA core: one wave computes a 64(M) x 32(N) fp32 tile of  C = A * W^T.
//   A: (64 x K) row-major with leading dim lda (k-contiguous)
//   W: (N x K)  row-major (k-contiguous)  -> NT GEMM, both operands k-major.
// A-frag (16x4, f32): lane l(0..15), half h: float2 at A[(mt*16+l)*lda + k0+2h]
// B-frag (4x16):                      float2 at W[(n0+nt*16+l)*K  + k0+2h]
// ---------------------------------------------------------------------------
__device__ __forceinline__ void wmma_tile_64x32(
    const float* __restrict__ A, int lda,
    const float* __restrict__ W, int K,
    int n0, int l, int half, v8f acc[4][2])
{
    const float* aBase = A + (size_t)l * lda + 2 * half;
    const float* w0 = W + (size_t)(n0 + l) * K + 2 * half;
    const float* w1 = W + (size_t)(n0 + 16 + l) * K + 2 * half;

    for (int k0 = 0; k0 < K; k0 += 4) {
        v2f b0 = *(const v2f*)(w0 + k0);
        v2f b1 = *(const v2f*)(w1 + k0);
#pragma unroll
        for (int mt = 0; mt < 4; ++mt) {
            v2f a = *(const v2f*)(aBase + (size_t)mt * 16 * lda + k0);
            acc[mt][0] = __builtin_amdgcn_wmma_f32_16x16x4_f32(
                false, a, false, b0, (short)0, acc[mt][0], false, false);
            acc[mt][1] = __builtin_amdgcn_wmma_f32_16x16x4_f32(
                false, a, false, b1, (short)0, acc[mt][1], false, false);
        }
    }
}

// ---------------- gates GEMM: gates(64x2048) = xh(64x1024) @ Wcat^T + bcat --
__global__ void gemm_gates_kernel(const float* __restrict__ xh,
                                  const float* __restrict__ Wcat,
                                  const float* __restrict__ bcat,
                                  float* __restrict__ gates)
{
    const int wave = blockIdx.x * (blockDim.x >> 5) + (threadIdx.x >> 5);
    const int lane = threadIdx.x & 31;
    const int half = lane >> 4;
    const int l    = lane & 15;
    const int n0   = wave * 32;                      // 64 waves cover N=2048

    v8f acc[4][2] = {};
    wmma_tile_64x32(xh, KCAT, Wcat, KCAT, n0, l, half, acc);

#pragma unroll
    for (int mt = 0; mt < 4; ++mt)
#pragma unroll
        for (int nt = 0; nt < 2; ++nt) {
            const int n = n0 + nt * 16 + l;
            const float bias = bcat[n];
#pragma unroll
            for (int r = 0; r < 8; ++r) {
                const int m = mt * 16 + half * 8 + r;
                gates[(size_t)m * G4H + n] = acc[mt][nt][r] + bias;
            }
        }
}

// -------- logits GEMM: out[b,t,:] = done ? 0 : h(64x512) @ Wlin^T + blin ----
__global__ void gemm_logits_kernel(const float* __restrict__ xh,   // h at +512
                                   const float* __restrict__ Wlin,
                                   const float* __restrict__ blin,
                                   const int* __restrict__ done,
                                   float* __restrict__ out, int t)
{
    const int wave = blockIdx.x * (blockDim.x >> 5) + (threadIdx.x >> 5);
    const int lane = threadIdx.x & 31;
    const int half = lane >> 4;
    const int l    = lane & 15;
    const int n0   = wave * 32;                      // 1000 waves cover N=32000

    v8f acc[4][2] = {};
    wmma_tile_64x32(xh + HID, KCAT, Wlin, HID, n0, l, half, acc);

    const bool dn = (*done != 0);                    // uniform per step
#pragma unroll
    for (int mt = 0; mt < 4; ++mt)
#pragma unroll
        for (int nt = 0; nt < 2; ++nt) {
            const int n = n0 + nt * 16 + l;
            const float bias = blin[n];
#pragma unroll
            for (int r = 0; r < 8; ++r) {
                const int m = mt * 16 + half * 8 + r;        // batch row
                const float v = dn ? 0.0f : (acc[mt][nt][r] + bias);
                out[(size_t)m * OUTROW + (size_t)t * VOCAB + n] = v;
            }
        }
}

// ---------------- one-time weight packing: Wcat = [W_ih | W_hh], bcat -------
__global__ void pack_kernel(const float* __restrict__ W_ih,
                            const float* __restrict__ W_hh,
                            const float* __restrict__ b_ih,
                            const float* __restrict__ b_hh,
                            float* __restrict__ Wcat,
                            float* __restrict__ bcat)
{
    const int tid = blockIdx.x * blockDim.x + threadIdx.x;   // 2048*1024 total
    const int n = tid >> 10;
    const int k = tid & 1023;
    Wcat[tid] = (k < HID) ? W_ih[(size_t)n * HID + k]
                          : W_hh[(size_t)n * HID + (k - HID)];
    if (tid < G4H) bcat[tid] = b_ih[tid] + b_hh[tid];
}

// ---------------- state init: h0=features, c0=0, word=START, flags=0 --------
__global__ void init_kernel(const float* __restrict__ features,
                            float* __restrict__ xh, float* __restrict__ c,
                            int* __restrict__ word, int* __restrict__ end_mask,
                            int* __restrict__ done)
{
    const int tid = blockIdx.x * blockDim.x + threadIdx.x;   // 64*512 total
    const int b = tid >> 9;
    const int k = tid & 511;
    xh[(size_t)b * KCAT + HID + k] = features[tid];          // h0 = features
    c[tid] = 0.0f;
    if (tid < B) { word[tid] = 1; end_mask[tid] = 0; }       // START_TOK = 1
    if (tid == 0) *done = 0;
}

// ---------------- embedding gather: xh[:,0:512] = embedding[word] -----------
__global__ void gather_kernel(const float* __restrict__ emb,
                              const int* __restrict__ word,
                              float* __restrict__ xh)
{
    const int tid = blockIdx.x * blockDim.x + threadIdx.x;   // 64*512 total
    const int b = tid >> 9;
    const int k = tid & 511;
    xh[(size_t)b * KCAT + k] = emb[(size_t)word[b] * EMB + k];
}

// ---------------- LSTM cell pointwise ---------------------------------------
__device__ __forceinline__ float sigmoidf(float x) { return 1.0f / (1.0f + expf(-x)); }

__global__ void cell_kernel(const float* __restrict__ gates,
                            float* __restrict__ c, float* __restrict__ xh)
{
    const int tid = blockIdx.x * blockDim.x + threadIdx.x;   // 64*512 total
    const int b = tid >> 9;
    const int k = tid & 511;
    const float* g = gates + (size_t)b * G4H;
    const float i = sigmoidf(g[k]);
    const float f = sigmoidf(g[HID + k]);
    const float gg = tanhf(g[2 * HID + k]);
    const float o = sigmoidf(g[3 * HID + k]);
    const float cc = f * c[tid] + i * gg;
    c[tid] = cc;
    xh[(size_t)b * KCAT + HID + k] = o * tanhf(cc);          // h -> xh[:,512:]
}

// ---------------- per-row argmax over VOCAB (first-max tie-break) -----------
__global__ void argmax_kernel(const float* __restrict__ out, int t,
                              int* __restrict__ word, int* __restrict__ end_mask)
{
    const int b = blockIdx.x;
    const float* row = out + (size_t)b * OUTROW + (size_t)t * VOCAB;

    float best = -INFINITY;
    int bi = 0;
    for (int n = threadIdx.x; n < VOCAB; n += 256) {
        const float v = row[n];
        if (v > best) { best = v; bi = n; }                  // first occurrence
    }
    __shared__ float sv[256];
    __shared__ int   si[256];
    sv[threadIdx.x] = best; si[threadIdx.x] = bi;
    __syncthreads();
    for (int s = 128; s > 0; s >>= 1) {
        if (threadIdx.x < s) {
            const float v2 = sv[threadIdx.x + s];
            const int   i2 = si[threadIdx.x + s];
            if (v2 > sv[threadIdx.x] ||
                (v2 == sv[threadIdx.x] && i2 < si[threadIdx.x])) {
                sv[threadIdx.x] = v2; si[threadIdx.x] = i2;
            }
        }
        __syncthreads();
    }
    if (threadIdx.x == 0) {
        word[b] = si[0];
        if (si[0] == 2) end_mask[b] = 1;                     // END_TOK = 2
    }
}

// ---------------- done |= all(end_mask) -------------------------------------
__global__ void done_kernel(const int* __restrict__ end_mask, int* __restrict__ done)
{
    __shared__ int notall;
    if (threadIdx.x == 0) notall = 0;
    __syncthreads();
    if (threadIdx.x < B && end_mask[threadIdx.x] == 0) atomicOr(&notall, 1);
    __syncthreads();
    if (threadIdx.x == 0 && notall == 0) *done = 1;          // sticky within a launch
}

// ---------------------------------------------------------------------------
extern "C" void kernel_launch(void* const* d_in, const int* in_sizes, int n_in,
                              void* d_out, int out_size, void* d_ws, size_t ws_size,
                              hipStream_t stream)
{
    (void)in_sizes; (void)n_in; (void)out_size; (void)ws_size;

    const float* features  = (const float*)d_in[0];
    const float* embedding = (const float*)d_in[1];
    const float* W_ih      = (const float*)d_in[2];
    const float* W_hh      = (const float*)d_in[3];
    const float* b_ih      = (const float*)d_in[4];
    const float* b_hh      = (const float*)d_in[5];
    const float* W_lin     = (const float*)d_in[6];
    const float* b_lin     = (const float*)d_in[7];
    float* out = (float*)d_out;

    // Workspace layout (~9.3 MB)
    float* Wcat     = (float*)d_ws;               // 2048*1024
    float* gates    = Wcat + (size_t)G4H * KCAT;  // 64*2048
    float* xh       = gates + (size_t)B * G4H;    // 64*1024  [x | h]
    float* c        = xh + (size_t)B * KCAT;      // 64*512
    float* bcat     = c + (size_t)B * HID;        // 2048
    int*   word     = (int*)(bcat + G4H);         // 64
    int*   end_mask = word + B;                   // 64
    int*   done     = end_mask + B;               // 1

    init_kernel<<<(B * HID) / 256, 256, 0, stream>>>(features, xh, c, word, end_mask, done);
    pack_kernel<<<(G4H * KCAT) / 256, 256, 0, stream>>>(W_ih, W_hh, b_ih, b_hh, Wcat, bcat);

    for (int t = 0; t < STEPS; ++t) {
        gather_kernel<<<(B * EMB) / 256, 256, 0, stream>>>(embedding, word, xh);
        // 64 waves (8 blocks x 8 waves) cover N=2048
        gemm_gates_kernel<<<(G4H / 32) / 8, 256, 0, stream>>>(xh, Wcat, bcat, gates);
        cell_kernel<<<(B * HID) / 256, 256, 0, stream>>>(gates, c, xh);
        // 1000 waves (125 blocks x 8 waves) cover N=32000
        gemm_logits_kernel<<<(VOCAB / 32) / 8, 256, 0, stream>>>(xh, W_lin, b_lin, done, out, t);
        argmax_kernel<<<B, 256, 0, stream>>>(out, t, word, end_mask);
        done_kernel<<<1, 64, 0, stream>>>(end_mask, done);
    }
}